// LightGCN_82420422410784
// MI455X (gfx1250) — compile-verified
//
#include <hip/hip_runtime.h>
#include <hip/hip_bf16.h>

// ---------------------------------------------------------------------------
// LightGCN forward for MI455X (gfx1250, wave32).
//  - SpMM layers: edge-parallel scatter with hardware global_atomic_add_f32;
//    both ping-pong feature buffers (76.8 MB each) are L2-resident (192 MB L2).
//  - Epilogue: batched dot products via V_WMMA_F32_16X16X4_F32 Gram tiles
//    (full fp32 precision), diagonal extraction per the ISA C/D layout.
// ---------------------------------------------------------------------------

#define EMBED 128

typedef __attribute__((ext_vector_type(2))) float v2f;
typedef __attribute__((ext_vector_type(8))) float v8f;

// acc = hA = feats (concat of user/item embeddings); hB = 0
__global__ void lgcn_init(const float* __restrict__ uEmbd,
                          const float* __restrict__ iEmbd,
                          float* __restrict__ hA,
                          float* __restrict__ hB,
                          float* __restrict__ acc,
                          int userNum, long total4) {
    long i = (long)blockIdx.x * blockDim.x + threadIdx.x;
    if (i >= total4) return;
    long g    = i * 4;              // flat float index
    long node = g >> 7;             // / EMBED
    int  off  = (int)(g & 127);     // % EMBED
    const float* src = (node < userNum)
        ? (uEmbd + node * (long)EMBED + off)
        : (iEmbd + (node - userNum) * (long)EMBED + off);
    float4 f = *(const float4*)src;
    ((float4*)hA)[i]  = f;
    ((float4*)acc)[i] = f;
    ((float4*)hB)[i]  = make_float4(0.f, 0.f, 0.f, 0.f);
}

// One wave per group of EPW edges: hn[row[e]] += val[e] * h[col[e]]
// Each lane owns 4 consecutive feature columns -> float4 gather + 4 fp32 atomics.
#define EPW 4
__global__ void spmm_scatter(const float* __restrict__ h,
                             float* __restrict__ hn,
                             const float* __restrict__ val,
                             const int* __restrict__ row,
                             const int* __restrict__ col,
                             int nnz) {
    const int lane = threadIdx.x & 31;
    const long wave = ((long)blockIdx.x * blockDim.x + threadIdx.x) >> 5;
    long e0 = wave * EPW;
    if (e0 >= nnz) return;
    long rem = (long)nnz - e0;
    int  n   = rem < EPW ? (int)rem : EPW;

    // Prefetch gathered source rows (global_prefetch_b8) to cover L2 latency.
    for (int i = 0; i < n; ++i) {
        int c = col[e0 + i];
        __builtin_prefetch(h + (size_t)c * EMBED + lane * 4, 0, 0);
    }
    for (int i = 0; i < n; ++i) {
        long e = e0 + i;
        int   r = row[e];
        int   c = col[e];
        float w = val[e];
        float4 s = ((const float4*)(h + (size_t)c * EMBED))[lane];
        float* dst = hn + (size_t)r * EMBED + (size_t)lane * 4;
        unsafeAtomicAdd(dst + 0, w * s.x);
        unsafeAtomicAdd(dst + 1, w * s.y);
        unsafeAtomicAdd(dst + 2, w * s.z);
        unsafeAtomicAdd(dst + 3, w * s.w);
    }
}

// acc += src; tozero = 0 (prepares the next layer's destination buffer)
__global__ void acc_add_zero(float* __restrict__ acc,
                             const float* __restrict__ src,
                             float* __restrict__ tozero,
                             long total4) {
    long i = (long)blockIdx.x * blockDim.x + threadIdx.x;
    if (i >= total4) return;
    float4 a = ((const float4*)acc)[i];
    float4 s = ((const float4*)src)[i];
    a.x += s.x; a.y += s.y; a.z += s.z; a.w += s.w;
    ((float4*)acc)[i]   = a;
    ((float4*)tozero)[i] = make_float4(0.f, 0.f, 0.f, 0.f);
}

// One wave per 16 batch rows. G = U * V^T via 32 chained WMMA f32 16x16x4,
// out = diag(G) / 16  (folds final = acc/4 on both factors).
__global__ void __launch_bounds__(32)
gram_dot(const float* __restrict__ acc,
         const int* __restrict__ uIdx,
         const int* __restrict__ iIdx,
         float* __restrict__ out,
         int userNum) {
    const int tile = blockIdx.x;          // 16 batch rows
    const int lane = threadIdx.x;         // 0..31, EXEC all ones (required by WMMA)
    const int m    = lane & 15;           // matrix row this lane feeds
    const int half = lane >> 4;           // K sub-column selector

    const int b = tile * 16 + m;
    const float* urow = acc + (size_t)uIdx[b] * EMBED;
    const float* vrow = acc + ((size_t)iIdx[b] + (size_t)userNum) * EMBED;

    v8f c = {0.f, 0.f, 0.f, 0.f, 0.f, 0.f, 0.f, 0.f};
    #pragma unroll
    for (int k0 = 0; k0 < EMBED; k0 += 4) {
        const int kk = k0 + 2 * half;
        // A (16x4, M x K): lane l holds U[l%16][k0 + 2*(l/16) + {0,1}]
        v2f a;  a.x  = urow[kk]; a.y  = urow[kk + 1];
        // B (4x16, K x N) with B[k][n] = V[n][k]: identical per-lane pattern
        v2f bb; bb.x = vrow[kk]; bb.y = vrow[kk + 1];
        c = __builtin_amdgcn_wmma_f32_16x16x4_f32(
                /*neg_a=*/false, a, /*neg_b=*/false, bb,
                /*c_mod=*/(short)0, c, /*reuse_a=*/false, /*reuse_b=*/false);
    }
    // Diagonal: D[m][m] -> (m<8): VGPR m, lane m ; (m>=8): VGPR m-8, lane m+16.
    const float scale = 1.0f / 16.0f;     // (1/4) * (1/4)
    if (lane < 8) {
        out[tile * 16 + lane] = c[lane] * scale;
    } else if (lane >= 24) {
        out[tile * 16 + (lane - 16)] = c[lane - 24] * scale;
    }
}

extern "C" void kernel_launch(void* const* d_in, const int* in_sizes, int n_in,
                              void* d_out, int out_size, void* d_ws, size_t ws_size,
                              hipStream_t stream) {
    const float* uEmbd = (const float*)d_in[0];
    const float* iEmbd = (const float*)d_in[1];
    const float* Lval  = (const float*)d_in[2];
    const int*   Lrow  = (const int*)d_in[3];
    const int*   Lcol  = (const int*)d_in[4];
    const int*   uIdx  = (const int*)d_in[5];
    const int*   iIdx  = (const int*)d_in[6];

    const int userNum = in_sizes[0] / EMBED;
    const int itemNum = in_sizes[1] / EMBED;
    const int N       = userNum + itemNum;
    const int nnz     = in_sizes[2];
    const int batch   = in_sizes[5];

    const size_t S = (size_t)N * EMBED;   // floats per feature buffer
    float* hA  = (float*)d_ws;
    float* hB  = hA + S;
    float* acc = hB + S;

    const long total4 = (long)S / 4;      // float4 elements
    const int  TB = 256;
    const int  gridElem = (int)((total4 + TB - 1) / TB);

    const long waves   = ((long)nnz + EPW - 1) / EPW;
    const int  gridSc  = (int)((waves * 32 + TB - 1) / TB);

    // acc = hA = feats ; hB = 0
    lgcn_init<<<gridElem, TB, 0, stream>>>(uEmbd, iEmbd, hA, hB, acc, userNum, total4);

    // Layer 1: hB = L*hA ; acc += hB ; hA = 0
    spmm_scatter<<<gridSc, TB, 0, stream>>>(hA, hB, Lval, Lrow, Lcol, nnz);
    acc_add_zero<<<gridElem, TB, 0, stream>>>(acc, hB, hA, total4);

    // Layer 2: hA = L*hB ; acc += hA ; hB = 0
    spmm_scatter<<<gridSc, TB, 0, stream>>>(hB, hA, Lval, Lrow, Lcol, nnz);
    acc_add_zero<<<gridElem, TB, 0, stream>>>(acc, hA, hB, total4);

    // Layer 3: hB = L*hA ; acc += hB (re-zero hA, harmless)
    spmm_scatter<<<gridSc, TB, 0, stream>>>(hA, hB, Lval, Lrow, Lcol, nnz);
    acc_add_zero<<<gridElem, TB, 0, stream>>>(acc, hB, hA, total4);

    // Epilogue: WMMA Gram-tile dot products, out[b] = <acc_u, acc_v>/16
    gram_dot<<<batch / 16, 32, 0, stream>>>(acc, uIdx, iIdx, (float*)d_out, userNum);
}